// TradingSignalLoss_38087769980986
// MI455X (gfx1250) — compile-verified
//
#include <hip/hip_runtime.h>

typedef __attribute__((ext_vector_type(2))) float v2f;
typedef __attribute__((ext_vector_type(8))) float v8f;

#define BATCH_N 8388608
#define NGROUP  (BATCH_N / 4)   // 4 rows per thread-iteration (3x float4 logits, 1x int4 targets)

__global__ void tsl_init_ws(float* ws) { ws[0] = 0.0f; }

__global__ __launch_bounds__(256) void tsl_main(const float* __restrict__ logits,
                                                const int*   __restrict__ targets,
                                                const float* __restrict__ w,
                                                float* __restrict__ ws) {
    __shared__ float partials[256];

    const float w0 = w[0], w1 = w[1], w2 = w[2];

    float acc = 0.0f;
    const int stride = gridDim.x * blockDim.x;
    for (int g = blockIdx.x * blockDim.x + threadIdx.x; g < NGROUP; g += stride) {
        // 4 rows = 12 floats = 3 x b128 loads, 16B-aligned (48B stride)
        const float4* lp = (const float4*)(logits + (size_t)g * 12);
        float4 va = lp[0];
        float4 vb = lp[1];
        float4 vc = lp[2];
        int4 t4 = ((const int4*)targets)[g];

        float rl[4][3] = {{va.x, va.y, va.z},
                          {va.w, vb.x, vb.y},
                          {vb.z, vb.w, vc.x},
                          {vc.y, vc.z, vc.w}};
        int tt[4] = {t4.x, t4.y, t4.z, t4.w};

#pragma unroll
        for (int r = 0; r < 4; ++r) {
            float l0 = rl[r][0], l1 = rl[r][1], l2 = rl[r][2];
            int t = tt[r];

            // argmax (first-max tie semantics, matches jnp.argmax)
            int pred = 0; float m = l0;
            if (l1 > m) { m = l1; pred = 1; }
            if (l2 > m) { m = l2; pred = 2; }

            // stable softmax pieces; e[pred] == 1 so max_prob = 1/s
            float e0 = __expf(l0 - m);
            float e1 = __expf(l1 - m);
            float e2 = __expf(l2 - m);
            float s = e0 + e1 + e2;
            float inv_s = __builtin_amdgcn_rcpf(s);

            float lt = (t == 0) ? l0 : ((t == 1) ? l1 : l2);
            float et = (t == 0) ? e0 : ((t == 1) ? e1 : e2);

            float ce = __logf(s) - (lt - m);     // cross-entropy
            float pt = et * inv_s;               // exp(-ce)
            float omp = 1.0f - pt;
            float alpha = (t == 0) ? w0 : ((t == 1) ? w1 : w2);

            float contrib = alpha * omp * omp * ce;        // focal term

            // PENALTY_MATRIX[pred][tgt] * 1.5 == {0,0.5,1.5}[|pred-tgt|] * 1.5
            int d = pred - t; d = (d < 0) ? -d : d;
            contrib += (d == 1) ? 0.75f : ((d == 2) ? 2.25f : 0.0f);

            // confidence term: 0.1 * (1-correct) * max_prob, max_prob = 1/s
            if (pred != t) contrib += 0.1f * inv_s;

            acc += contrib;
        }
    }

    partials[threadIdx.x] = acc;
    __syncthreads();

    // Block reduction on the matrix unit: D = Ones(16x4) x B(4x16) + C.
    // Each WMMA folds 64 partials into 16 column sums; chain 4x over 256.
    if (threadIdx.x < 32) {            // whole wave 0 -> EXEC all-ones inside
        int lane = threadIdx.x;
        v2f ones; ones.x = 1.0f; ones.y = 1.0f;
        v8f dmat = {};
#pragma unroll
        for (int i = 0; i < 4; ++i) {
            v2f bmat;
            bmat.x = partials[i * 64 + lane];
            bmat.y = partials[i * 64 + 32 + lane];
            dmat = __builtin_amdgcn_wmma_f32_16x16x4_f32(
                false, ones, false, bmat, (short)0, dmat, false, false);
        }
        // D row 0 lives in VGPR0 across lanes 0-15 (lanes 16-31 hold the
        // identical M=8 row). Sum the 16 column sums with xor shuffles.
        float v = dmat[0];
        v += __shfl_xor(v, 8, 32);
        v += __shfl_xor(v, 4, 32);
        v += __shfl_xor(v, 2, 32);
        v += __shfl_xor(v, 1, 32);
        if (lane == 0) atomicAdd(ws, v);
    }
}

__global__ void tsl_finalize(const float* __restrict__ ws, float* __restrict__ out) {
    out[0] = ws[0] * (1.0f / (float)BATCH_N);
}

extern "C" void kernel_launch(void* const* d_in, const int* in_sizes, int n_in,
                              void* d_out, int out_size, void* d_ws, size_t ws_size,
                              hipStream_t stream) {
    const float* logits  = (const float*)d_in[0];
    const int*   targets = (const int*)d_in[1];
    const float* weights = (const float*)d_in[2];
    float* out = (float*)d_out;
    float* ws  = (float*)d_ws;

    tsl_init_ws<<<1, 1, 0, stream>>>(ws);

    // 8192 blocks x 256 threads = 2,097,152 threads: exactly one 4-row group
    // per thread (64B of traffic each) -> max memory-level parallelism.
    tsl_main<<<8192, 256, 0, stream>>>(logits, targets, weights, ws);

    tsl_finalize<<<1, 1, 0, stream>>>(ws, out);
}